// CompositeLoss_420906795783
// MI455X (gfx1250) — compile-verified
//
#include <hip/hip_runtime.h>
#include <math.h>

// Problem constants (fixed by the reference harness)
#define BB 16
#define HH 512
#define WW 512
#define NN (BB * HH * WW)   // 4,194,304
#define BIGF 1.0e6f

typedef __attribute__((ext_vector_type(2))) float v2f;
typedef __attribute__((ext_vector_type(8))) float v8f;

// ---------------------------------------------------------------------------
// Wave32 sum-reduction using V_WMMA_F32_16X16X4_F32 (fp32-exact):
//  A(16x4): lane m (0..15) VGPR0 holds K=0, lanes 16..31 VGPR0 holds K=2.
//  With a = {x, 0}, B = ones:  D[m,n] = x[m] + x[m+16] for every n.
//  D VGPR v: lanes 0..15 -> M=v, lanes 16..31 -> M=v+8.
//  Sum of the 8 D VGPRs gives half-sums; one shfl_xor(16) completes it.
// Requires EXEC all-ones: call with full, convergent waves only.
// ---------------------------------------------------------------------------
__device__ __forceinline__ float wave_reduce_sum(float x) {
#if __has_builtin(__builtin_amdgcn_wmma_f32_16x16x4_f32)
  v2f a; a.x = x; a.y = 0.0f;
  v2f b; b.x = 1.0f; b.y = 1.0f;
  v8f c = {};
  c = __builtin_amdgcn_wmma_f32_16x16x4_f32(false, a, false, b, (short)0, c,
                                            false, false);
  float s = c[0] + c[1] + c[2] + c[3] + c[4] + c[5] + c[6] + c[7];
  s += __shfl_xor(s, 16, 32);
  return s;
#else
  float s = x;
  s += __shfl_xor(s, 1, 32);
  s += __shfl_xor(s, 2, 32);
  s += __shfl_xor(s, 4, 32);
  s += __shfl_xor(s, 8, 32);
  s += __shfl_xor(s, 16, 32);
  return s;
#endif
}

// Deterministic block (256-thread / 8-wave) reduction; result in all threads.
__device__ __forceinline__ float block_reduce_sum(float x, float* s_tmp) {
  float w = wave_reduce_sum(x);
  const int wid  = threadIdx.x >> 5;
  const int lane = threadIdx.x & 31;
  if (lane == 0) s_tmp[wid] = w;
  __syncthreads();
  float r = 0.0f;
#pragma unroll
  for (int i = 0; i < 8; ++i) r += s_tmp[i];   // fixed order -> deterministic
  __syncthreads();
  return r;
}

// ---------------------------------------------------------------------------
// Kernel 1: row pass of the EDT. One thread per row (B*H = 8192 rows).
// Writes h[b,i,j] = min(g,BIG)^2 + i*i  (g = 1-D distance along the row).
// ---------------------------------------------------------------------------
__global__ void edt_rows_kernel(const int* __restrict__ tgt,
                                float* __restrict__ h) {
  const int row  = blockIdx.x * blockDim.x + threadIdx.x;  // 0..B*H-1
  const int base = row * WW;
  const int i    = row & (HH - 1);
  const float ii = (float)i * (float)i;

  float left = -BIGF;
  for (int j = 0; j < WW; ++j) {
    const float pos = (float)j;
    if (tgt[base + j] > 0) left = pos;
    h[base + j] = pos - left;            // stash left-distance
  }
  float right = BIGF;
  for (int j = WW - 1; j >= 0; --j) {
    const float pos = (float)j;
    if (tgt[base + j] > 0) right = pos;
    float g = fminf(h[base + j], right - pos);
    g = fminf(g, BIGF);
    h[base + j] = g * g + ii;            // h = g2 + i^2
  }
}

// ---------------------------------------------------------------------------
// Kernel 2: column min-plus (exact, matches reference O(H^2) scan) fused with
// the boundary-loss accumulation. One workgroup per (image, 32-column tile).
//  d2[i] = i^2 + min_{i'} (h[i'] - 2*i*i')   -- integer-exact in fp32 here.
// ---------------------------------------------------------------------------
__global__ void __launch_bounds__(256)
edt_cols_boundary_kernel(const float* __restrict__ h,
                         const float* __restrict__ logits,
                         const int* __restrict__ tgt,
                         float* __restrict__ bl_part) {
  __shared__ float s_h[HH * 32];         // 64 KB tile, reused for reduction

  const int b    = blockIdx.x >> 4;      // 16 tiles per image
  const int tile = blockIdx.x & 15;
  const int j0   = tile * 32;

  // Coalesced tile load: h[b, 0..511, j0..j0+31]
  const int ibase = b * HH;
  for (int idx = threadIdx.x; idx < HH * 32; idx += 256) {
    const int r = idx >> 5;
    const int c = idx & 31;
    s_h[idx] = h[(ibase + r) * WW + j0 + c];
  }
  __syncthreads();

  const int jt = threadIdx.x & 31;       // column within tile (lane id)
  const int ig = threadIdx.x >> 5;       // 0..7: i-phase
  const int j  = j0 + jt;

  float bsum = 0.0f;
  for (int kb = 0; kb < 8; ++kb) {
    float m[8], cf[8], iv[8];
#pragma unroll
    for (int u = 0; u < 8; ++u) {
      const int i = ig + 8 * (kb * 8 + u);
      iv[u] = (float)i;
      cf[u] = -2.0f * iv[u];
      m[u]  = 3.0e38f;
    }
    for (int ip = 0; ip < HH; ++ip) {
      const float v   = s_h[ip * 32 + jt];   // conflict-free: lanes->banks
      const float ipf = (float)ip;
#pragma unroll
      for (int u = 0; u < 8; ++u)
        m[u] = fminf(m[u], fmaf(cf[u], ipf, v));
    }
#pragma unroll
    for (int u = 0; u < 8; ++u) {
      const int i = ig + 8 * (kb * 8 + u);
      const float d2 = m[u] + iv[u] * iv[u];
      const float d  = sqrtf(d2);
      const int gi   = (ibase + i) * WW + j;  // coalesced across the wave
      const float z  = logits[gi];
      const float gf = (tgt[gi] > 0) ? 1.0f : 0.0f;
      const float p  = 1.0f / (1.0f + __expf(-z));
      bsum += fabsf(p - gf) * d;
    }
  }

  __syncthreads();                       // done reading s_h; reuse as scratch
  const float tot = block_reduce_sum(bsum, s_h);
  if (threadIdx.x == 0) bl_part[blockIdx.x] = tot;
}

// ---------------------------------------------------------------------------
// Kernel 3: elementwise partials (TP, sumP, sumG, BCE). 256 blocks / sample
// so per-block partials can be regrouped per-sample for Dice.
// ---------------------------------------------------------------------------
__global__ void __launch_bounds__(256)
elemwise_partials_kernel(const float* __restrict__ logits,
                         const int* __restrict__ tgt,
                         float* __restrict__ part) {
  __shared__ float s_tmp[8];
  const int base = (blockIdx.x * 256 + threadIdx.x) * 4;

  const float4 z4 = *(const float4*)(logits + base);
  const int4   t4 = *(const int4*)(tgt + base);

  float tp = 0.0f, sp = 0.0f, sg = 0.0f, bce = 0.0f;
  const float zv[4] = {z4.x, z4.y, z4.z, z4.w};
  const int   tv[4] = {t4.x, t4.y, t4.z, t4.w};
#pragma unroll
  for (int k = 0; k < 4; ++k) {
    const float z  = zv[k];
    const float gf = (tv[k] > 0) ? 1.0f : 0.0f;
    const float p  = 1.0f / (1.0f + __expf(-z));
    tp  += p * gf;
    sp  += p;
    sg  += gf;
    bce += fmaxf(z, 0.0f) - z * gf + log1pf(__expf(-fabsf(z)));
  }

  tp  = block_reduce_sum(tp, s_tmp);
  sp  = block_reduce_sum(sp, s_tmp);
  sg  = block_reduce_sum(sg, s_tmp);
  bce = block_reduce_sum(bce, s_tmp);
  if (threadIdx.x == 0) {
    float* o = part + blockIdx.x * 4;
    o[0] = tp; o[1] = sp; o[2] = sg; o[3] = bce;
  }
}

// ---------------------------------------------------------------------------
// Kernel 4: final deterministic combine -> scalar loss.
// ---------------------------------------------------------------------------
__global__ void __launch_bounds__(256)
finalize_kernel(const float* __restrict__ bl_part,   // 256
                const float* __restrict__ k3_part,   // 4096 * 4
                float* __restrict__ out) {
  __shared__ float s_tmp[8];
  const int t = threadIdx.x;

  // Boundary loss
  const float bl_sum = block_reduce_sum(bl_part[t], s_tmp);

  // Global sums (fixed strided order per thread -> deterministic)
  float tp = 0.0f, sp = 0.0f, sg = 0.0f, bce = 0.0f;
  for (int idx = t; idx < 4096; idx += 256) {
    const float* o = k3_part + idx * 4;
    tp += o[0]; sp += o[1]; sg += o[2]; bce += o[3];
  }
  tp  = block_reduce_sum(tp, s_tmp);
  sp  = block_reduce_sum(sp, s_tmp);
  sg  = block_reduce_sum(sg, s_tmp);
  bce = block_reduce_sum(bce, s_tmp);

  // Per-sample Dice: 256 block-partials per sample
  float dice_acc = 0.0f;
  for (int s = 0; s < BB; ++s) {
    const float* o = k3_part + (s * 256 + t) * 4;
    const float I = block_reduce_sum(o[0], s_tmp);
    const float P = block_reduce_sum(o[1], s_tmp);
    const float G = block_reduce_sum(o[2], s_tmp);
    dice_acc += (2.0f * I + 1.0f) / (P + G + 1.0f);
  }

  if (t == 0) {
    const float Nf    = (float)NN;
    const float bl    = bl_sum / Nf;
    const float bce_m = bce / Nf;
    const float FPv   = sp - tp;
    const float FNv   = sg - tp;
    const float TI    = (tp + 1e-6f) / (tp + 0.3f * FPv + 0.7f * FNv + 1e-6f);
    const float ft    = powf(fmaxf(1.0f - TI, 0.0f), 0.75f);
    const float dice  = 1.0f - dice_acc / (float)BB;
    out[0] = 1.0f * ft + 0.5f * bl + 0.5f * bce_m + 0.5f * dice;
  }
}

// ---------------------------------------------------------------------------
extern "C" void kernel_launch(void* const* d_in, const int* in_sizes, int n_in,
                              void* d_out, int out_size, void* d_ws,
                              size_t ws_size, hipStream_t stream) {
  (void)in_sizes; (void)n_in; (void)out_size; (void)ws_size;
  const float* logits = (const float*)d_in[0];
  const int*   tgt    = (const int*)d_in[1];

  float* ws      = (float*)d_ws;
  float* h       = ws;                 // NN floats (16 MB)
  float* bl_part = ws + NN;            // 256 floats
  float* k3_part = bl_part + 256;      // 4096*4 floats

  edt_rows_kernel<<<BB * HH / 256, 256, 0, stream>>>(tgt, h);
  edt_cols_boundary_kernel<<<BB * (WW / 32), 256, 0, stream>>>(h, logits, tgt,
                                                               bl_part);
  elemwise_partials_kernel<<<NN / 1024, 256, 0, stream>>>(logits, tgt, k3_part);
  finalize_kernel<<<1, 256, 0, stream>>>(bl_part, k3_part, (float*)d_out);
}